// AGAIN_11244224381606
// MI455X (gfx1250) — compile-verified
//
#include <hip/hip_runtime.h>

// ---------------- problem constants (match reference) ----------------
#define N_NODES 20000
#define N_EDGES 320000
#define IN_F    256
#define ED_F    16
#define H_HEADS 8
#define C1_     64
#define C2_     25
#define HC1     512
#define HC2     200
#define THRESH_ 0.03f
#define ATT_NEG 0.2f
#define MLP_NEG 0.01f
#define SCALE_F (1.0f / 3.4011973816621555f)   /* 1/log(30) */
#define EPS_BN  1e-5f

typedef __attribute__((ext_vector_type(2))) float v2f;
typedef __attribute__((ext_vector_type(8))) float v8f;

// order-preserving float<->uint encoding for atomicMax on signed floats
static __device__ __forceinline__ unsigned encf(float f) {
  unsigned u = __float_as_uint(f);
  return (u & 0x80000000u) ? ~u : (u | 0x80000000u);
}
static __device__ __forceinline__ float decf(unsigned e) {
  unsigned u = (e & 0x80000000u) ? (e & 0x7FFFFFFFu) : ~e;
  return __uint_as_float(u);
}

// async Global -> LDS 16-byte copy (ASYNCcnt-tracked, no VGPR round trip)
static __device__ __forceinline__ void async_copy_b128(unsigned lds_off, const void* gptr) {
  asm volatile("global_load_async_to_lds_b128 %0, %1, off"
               :: "v"(lds_off), "v"((unsigned long long)(uintptr_t)gptr)
               : "memory");
}
// wait until all of this wave's async transfers are done
static __device__ __forceinline__ void wait_async0() {
  asm volatile("s_wait_asynccnt 0x0" ::: "memory");
}
// wait until <=4 outstanding: by in-order retirement this means the
// previous tile's 4 transfers have landed while the newest 4 stay in flight
static __device__ __forceinline__ void wait_async4() {
  asm volatile("s_wait_asynccnt 0x4" ::: "memory");
}

// =====================================================================
// WMMA f32 GEMM:  D[M,Nn] = act(A[M,K] @ B[K,Nn] + bias) (+ resid)
// block = 128 threads (4 waves), tile 64x64, K stepped by 16,
// V_WMMA_F32_16X16X4_F32 per 16x16x4 sub-op.
// Double-buffered LDS tiles staged with global_load_async_to_lds_b128,
// pipelined via the ASYNCcnt threshold wait.
// flags: bit0 = leaky-relu(0.01), bit1 = add residual (stride Nn)
// =====================================================================
#define TBM 64
#define TBN 64
#define A_STRIDE 20   // 16 + pad: rows 80B (16B aligned); gcd(20,64)=4 -> conflict-free A frags
#define B_STRIDE 68   // 64 + pad: rows 272B (16B aligned)
#define A_TILE_F (TBM * A_STRIDE)
#define B_TILE_F (16 * B_STRIDE)

__global__ __launch_bounds__(128) void gat_gemm_wmma_f32(
    const float* __restrict__ A, const float* __restrict__ B,
    const float* __restrict__ bias, const float* __restrict__ resid,
    float* __restrict__ D, int M, int K, int Nn, int flags)
{
  __shared__ __align__(16) float As[2 * A_TILE_F];  // [buf][m][k] row-major (padded)
  __shared__ __align__(16) float Bs[2 * B_TILE_F];  // [buf][k][n]            (padded)

  const int tid  = threadIdx.x;
  const int wave = tid >> 5;          // 0..3  -> M sub-tile
  const int lane = tid & 31;
  const int l15  = lane & 15;
  const int lhi  = lane >> 4;         // 0/1 (K pair select per ISA layout)
  const int mblk = blockIdx.y * TBM;
  const int nblk = blockIdx.x * TBN;

  // generic shared pointers: low 32 bits == LDS byte offset (ISA aperture rule)
  const unsigned asBase = (unsigned)(uintptr_t)(void*)As;
  const unsigned bsBase = (unsigned)(uintptr_t)(void*)Bs;

  // stage one 64x16 A tile + 16x64 B tile into LDS buffer `buf`
  // (exactly 4 async b128 per thread when the tile is interior)
  auto stage = [&](int k0, int buf) {
    const unsigned ab = asBase + (unsigned)(buf * A_TILE_F) * 4u;
    const unsigned bb = bsBase + (unsigned)(buf * B_TILE_F) * 4u;
    #pragma unroll
    for (int ii = 0; ii < 2; ++ii) {
      int i = tid + ii * 128;          // 0..255
      int r = i >> 2, c = (i & 3) * 4; // row, k-chunk (floats)
      int gm = mblk + r, gk = k0 + c;
      if (gm < M && gk < K) {
        async_copy_b128(ab + (unsigned)(r * A_STRIDE + c) * 4u, A + (long)gm * K + gk);
      } else {
        float* p = As + buf * A_TILE_F + r * A_STRIDE + c;
        p[0] = 0.f; p[1] = 0.f; p[2] = 0.f; p[3] = 0.f;
      }
    }
    #pragma unroll
    for (int ii = 0; ii < 2; ++ii) {
      int i = tid + ii * 128;
      int kk = i >> 4, c = (i & 15) * 4;
      int gk = k0 + kk, gn = nblk + c;
      if (gk < K && gn < Nn) {
        async_copy_b128(bb + (unsigned)(kk * B_STRIDE + c) * 4u, B + (long)gk * Nn + gn);
      } else {
        float* p = Bs + buf * B_TILE_F + kk * B_STRIDE + c;
        p[0] = 0.f; p[1] = 0.f; p[2] = 0.f; p[3] = 0.f;
      }
    }
  };
  // tile fully in-bounds -> every wave issues exactly 4 asyncs for it
  auto interior = [&](int k0) -> bool {
    return (mblk + TBM <= M) && (nblk + TBN <= Nn) && (k0 + 16 <= K);
  };

  v8f acc0 = {}; v8f acc1 = {}; v8f acc2 = {}; v8f acc3 = {};

  const int Kp = (K + 15) & ~15;
  stage(0, 0);                               // prologue: tile 0 -> buf 0
  for (int k0 = 0; k0 < Kp; k0 += 16) {
    const int buf = (k0 >> 4) & 1;
    const int k1 = k0 + 16;
    const bool hasNext = (k1 < Kp);
    if (hasNext) stage(k1, buf ^ 1);         // prefetch next tile into other buffer
    if (hasNext && interior(k1)) wait_async4();  // oldest tile drained, newest in flight
    else                         wait_async0();  // boundary: conservative drain
    __syncthreads();                         // tile k0 visible to all waves

    const float* Ab = As + buf * A_TILE_F;
    const float* Bb = Bs + buf * B_TILE_F;
    const int ar = (wave * 16 + l15) * A_STRIDE;   // A row base for this lane
    #pragma unroll
    for (int kk = 0; kk < 16; kk += 4) {
      const int ka = kk + lhi * 2;
      v2f a;
      a.x = Ab[ar + ka + 0];
      a.y = Ab[ar + ka + 1];
      const int br0 = (ka + 0) * B_STRIDE + l15;
      const int br1 = (ka + 1) * B_STRIDE + l15;
      v2f b0, b1, b2, b3;
      b0.x = Bb[br0 +  0];  b0.y = Bb[br1 +  0];
      b1.x = Bb[br0 + 16];  b1.y = Bb[br1 + 16];
      b2.x = Bb[br0 + 32];  b2.y = Bb[br1 + 32];
      b3.x = Bb[br0 + 48];  b3.y = Bb[br1 + 48];
      acc0 = __builtin_amdgcn_wmma_f32_16x16x4_f32(false, a, false, b0, (short)0, acc0, false, false);
      acc1 = __builtin_amdgcn_wmma_f32_16x16x4_f32(false, a, false, b1, (short)0, acc1, false, false);
      acc2 = __builtin_amdgcn_wmma_f32_16x16x4_f32(false, a, false, b2, (short)0, acc2, false, false);
      acc3 = __builtin_amdgcn_wmma_f32_16x16x4_f32(false, a, false, b3, (short)0, acc3, false, false);
    }
    __syncthreads();                         // protect this buffer before re-staging
  }

  // epilogue: C/D layout -> m = wave*16 + lhi*8 + j, n = ntile + l15
  v8f accs[4] = {acc0, acc1, acc2, acc3};
  #pragma unroll
  for (int t = 0; t < 4; ++t) {
    int gn = nblk + t * 16 + l15;
    if (gn >= Nn) continue;
    float bb = bias ? bias[gn] : 0.f;
    #pragma unroll
    for (int j = 0; j < 8; ++j) {
      int gm = mblk + wave * 16 + lhi * 8 + j;
      if (gm >= M) continue;
      float v = accs[t][j] + bb;
      if (flags & 1) v = (v >= 0.f) ? v : MLP_NEG * v;
      if (flags & 2) v += resid[(long)gm * Nn + gn];
      D[(long)gm * Nn + gn] = v;
    }
  }
}

// ---------------------------------------------------------------------
// fold We/be through ae per head:  wea[d][h] = sum_c We[d, h*C+c]*ae[h,c]
// ---------------------------------------------------------------------
__global__ void gat_wea_precompute(
    const float* __restrict__ We1, const float* __restrict__ ae1, const float* __restrict__ be1,
    const float* __restrict__ We2, const float* __restrict__ ae2, const float* __restrict__ be2,
    float* __restrict__ wea1, float* __restrict__ bea1,
    float* __restrict__ wea2, float* __restrict__ bea2)
{
  int t = threadIdx.x;
  if (t < 128) {
    int d = t >> 3, h = t & 7;
    float s = 0.f;
    for (int c = 0; c < C1_; ++c) s += We1[d * HC1 + h * C1_ + c] * ae1[h * C1_ + c];
    wea1[d * 8 + h] = s;
  } else {
    int i = t - 128;
    int d = i >> 3, h = i & 7;
    float s = 0.f;
    for (int c = 0; c < C2_; ++c) s += We2[d * HC2 + h * C2_ + c] * ae2[h * C2_ + c];
    wea2[d * 8 + h] = s;
  }
  if (t < 8) {
    float s = 0.f;
    for (int c = 0; c < C1_; ++c) s += be1[t * C1_ + c] * ae1[t * C1_ + c];
    bea1[t] = s;
  } else if (t < 16) {
    int h = t - 8;
    float s = 0.f;
    for (int c = 0; c < C2_; ++c) s += be2[h * C2_ + c] * ae2[h * C2_ + c];
    bea2[h] = s;
  }
}

// a_src[n,h] = sum_c h[n,h,c]*as[h,c]; a_dst likewise
__global__ void gat_node_att(const float* __restrict__ Hm,
                             const float* __restrict__ as_, const float* __restrict__ ad_,
                             float* __restrict__ asrc, float* __restrict__ adst,
                             int n, int C, int HC)
{
  int i = blockIdx.x * blockDim.x + threadIdx.x;
  if (i >= n * H_HEADS) return;
  int node = i >> 3, h = i & 7;
  const float* hp = Hm + (long)node * HC + h * C;
  const float* ap = as_ + h * C;
  const float* dp = ad_ + h * C;
  float s = 0.f, d = 0.f;
  for (int c = 0; c < C; ++c) { float v = hp[c]; s += v * ap[c]; d += v * dp[c]; }
  asrc[i] = s; adst[i] = d;
}

// logit[e,h] = lrelu(asrc[src]+adst[dst]+ea@wea+bea, 0.2)*SCALE ; segment max via atomicMax
__global__ void gat_edge_logit(const float* __restrict__ ea,
                               const int* __restrict__ src, const int* __restrict__ dst,
                               const float* __restrict__ asrc, const float* __restrict__ adst,
                               const float* __restrict__ wea, const float* __restrict__ bea,
                               float* __restrict__ logit, unsigned* __restrict__ mx, int Ecnt)
{
  int e = blockIdx.x * blockDim.x + threadIdx.x;
  if (e >= Ecnt) return;
  float eav[ED_F];
  #pragma unroll
  for (int d = 0; d < ED_F; ++d) eav[d] = ea[(long)e * ED_F + d];
  int s = src[e], t = dst[e];
  #pragma unroll
  for (int h = 0; h < H_HEADS; ++h) {
    float l = asrc[s * 8 + h] + adst[t * 8 + h] + bea[h];
    #pragma unroll
    for (int d = 0; d < ED_F; ++d) l += eav[d] * wea[d * 8 + h];
    l = (l >= 0.f) ? l : ATT_NEG * l;
    l *= SCALE_F;
    logit[(long)e * 8 + h] = l;
    atomicMax(&mx[t * 8 + h], encf(l));
  }
}

// ex = exp(logit - mx[dst]); den[dst] += ex  (ex overwrites logit in place)
__global__ void gat_edge_ex(float* __restrict__ logit, const int* __restrict__ dst,
                            const unsigned* __restrict__ mx, float* __restrict__ den, int Ecnt)
{
  int i = blockIdx.x * blockDim.x + threadIdx.x;
  if (i >= Ecnt * H_HEADS) return;
  int e = i >> 3, h = i & 7;
  int t = dst[e];
  float ex = expf(logit[i] - decf(mx[t * 8 + h]));
  logit[i] = ex;
  atomicAdd(&den[t * 8 + h], ex);
}

// alpha = ex/den, thresholded (in place)
__global__ void gat_edge_alpha(float* __restrict__ logit, const int* __restrict__ dst,
                               const float* __restrict__ den, int Ecnt)
{
  int i = blockIdx.x * blockDim.x + threadIdx.x;
  if (i >= Ecnt * H_HEADS) return;
  int e = i >> 3, h = i & 7;
  float a = logit[i] / den[dst[e] * 8 + h];
  logit[i] = (a < THRESH_) ? 0.f : a;
}

// out[dst, hc] += alpha[e, h] * h[src, hc]   (grid-stride over E*HC; L2-resident atomics)
__global__ void gat_scatter(const float* __restrict__ Hm, const float* __restrict__ alpha,
                            const int* __restrict__ src, const int* __restrict__ dst,
                            float* __restrict__ out, int Ecnt, int HC, int C)
{
  long total = (long)Ecnt * HC;
  for (long i = (long)blockIdx.x * blockDim.x + threadIdx.x; i < total;
       i += (long)gridDim.x * blockDim.x) {
    int e  = (int)(i / HC);
    int hc = (int)(i - (long)e * HC);
    int h  = hc / C;
    float a = alpha[(long)e * 8 + h];
    if (a != 0.f)
      atomicAdd(&out[(long)dst[e] * HC + hc], a * Hm[(long)src[e] * HC + hc]);
  }
}

__global__ void gat_bias_add(float* __restrict__ out, const float* __restrict__ b, long n, int HC)
{
  long i = (long)blockIdx.x * blockDim.x + threadIdx.x;
  if (i >= n) return;
  out[i] += b[i % HC];
}

__global__ void gat_fill_f32(float* __restrict__ p, float v, long n)
{
  long i = (long)blockIdx.x * blockDim.x + threadIdx.x;
  if (i < n) p[i] = v;
}

// per-column batch stats (biased variance), one block per column
__global__ void gat_bn_stats(const float* __restrict__ X, float* __restrict__ mean,
                             float* __restrict__ var, int M, int Nc)
{
  int col = blockIdx.x;
  float s = 0.f, s2 = 0.f;
  for (int r = threadIdx.x; r < M; r += blockDim.x) {
    float v = X[(long)r * Nc + col];
    s += v; s2 += v * v;
  }
  __shared__ float sh[256], sh2[256];
  sh[threadIdx.x] = s; sh2[threadIdx.x] = s2;
  __syncthreads();
  for (int off = blockDim.x >> 1; off > 0; off >>= 1) {
    if ((int)threadIdx.x < off) {
      sh[threadIdx.x]  += sh[threadIdx.x + off];
      sh2[threadIdx.x] += sh2[threadIdx.x + off];
    }
    __syncthreads();
  }
  if (threadIdx.x == 0) {
    float m = sh[0] / (float)M;
    mean[col] = m;
    var[col]  = sh2[0] / (float)M - m * m;
  }
}

__global__ void gat_bn_apply(float* __restrict__ X, const float* __restrict__ mean,
                             const float* __restrict__ var, const float* __restrict__ g,
                             const float* __restrict__ b, int M, int Nc)
{
  long i = (long)blockIdx.x * blockDim.x + threadIdx.x;
  if (i >= (long)M * Nc) return;
  int col = (int)(i % Nc);
  X[i] = g[col] * (X[i] - mean[col]) * rsqrtf(var[col] + EPS_BN) + b[col];
}

// final projection to scalar, one wave per row
__global__ void gat_fc5(const float* __restrict__ Hm, const float* __restrict__ w,
                        const float* __restrict__ b, float* __restrict__ out, int M, int K)
{
  int wave = threadIdx.x >> 5;
  int lane = threadIdx.x & 31;
  int n = blockIdx.x * (blockDim.x >> 5) + wave;
  if (n >= M) return;
  float s = 0.f;
  for (int k = lane; k < K; k += 32) s += Hm[(long)n * K + k] * w[k];
  for (int off = 16; off > 0; off >>= 1) s += __shfl_down(s, off, 32);
  if (lane == 0) out[n] = s + b[0];
}

// ---------------- workspace layout (float units) ----------------
static const size_t OFF_H1   = 0;                                 // N*512 (reused as h2: N*200)
static const size_t OFF_OUT  = OFF_H1   + (size_t)N_NODES * HC1;  // N*512 (conv outputs)
static const size_t OFF_TA   = OFF_OUT  + (size_t)N_NODES * HC1;  // N*256
static const size_t OFF_TB   = OFF_TA   + (size_t)N_NODES * IN_F; // N*256
static const size_t OFF_ASRC = OFF_TB   + (size_t)N_NODES * IN_F; // N*8
static const size_t OFF_ADST = OFF_ASRC + (size_t)N_NODES * 8;
static const size_t OFF_MX   = OFF_ADST + (size_t)N_NODES * 8;    // uint
static const size_t OFF_DEN  = OFF_MX   + (size_t)N_NODES * 8;
static const size_t OFF_ELOG = OFF_DEN  + (size_t)N_NODES * 8;    // E*8 (logit/ex/alpha)
static const size_t OFF_WEA1 = OFF_ELOG + (size_t)N_EDGES * 8;
static const size_t OFF_WEA2 = OFF_WEA1 + 128;
static const size_t OFF_BEA1 = OFF_WEA2 + 128;
static const size_t OFF_BEA2 = OFF_BEA1 + 8;
static const size_t OFF_BNM  = OFF_BEA2 + 8;
static const size_t OFF_BNV  = OFF_BNM  + IN_F;
// total ~33.92M floats (~136 MB)

static inline int cdivi(long a, long b) { return (int)((a + b - 1) / b); }

extern "C" void kernel_launch(void* const* d_in, const int* in_sizes, int n_in,
                              void* d_out, int out_size, void* d_ws, size_t ws_size,
                              hipStream_t stream)
{
  (void)in_sizes; (void)n_in; (void)out_size; (void)ws_size;

  const float* x   = (const float*)d_in[0];
  const int*   ei  = (const int*)d_in[1];
  const int*   src = ei;
  const int*   dst = ei + N_EDGES;
  const float* ea  = (const float*)d_in[2];

  const float* W1  = (const float*)d_in[3];
  const float* b1  = (const float*)d_in[4];
  const float* as1 = (const float*)d_in[5];
  const float* ad1 = (const float*)d_in[6];
  const float* We1 = (const float*)d_in[7];
  const float* be1 = (const float*)d_in[8];
  const float* ae1 = (const float*)d_in[9];
  const float* W2  = (const float*)d_in[10];
  const float* b2  = (const float*)d_in[11];
  const float* as2 = (const float*)d_in[12];
  const float* ad2 = (const float*)d_in[13];
  const float* We2 = (const float*)d_in[14];
  const float* be2 = (const float*)d_in[15];
  const float* ae2 = (const float*)d_in[16];
  const float* fc1_w = (const float*)d_in[17];
  const float* fc1_b = (const float*)d_in[18];
  const float* fc2_w = (const float*)d_in[19];
  const float* fc2_b = (const float*)d_in[20];
  const float* fc3_w = (const float*)d_in[21];
  const float* fc3_b = (const float*)d_in[22];
  const float* fc4_w = (const float*)d_in[23];
  const float* fc4_b = (const float*)d_in[24];
  const float* fc5_w = (const float*)d_in[25];
  const float* fc5_b = (const float*)d_in[26];
  const float* g1  = (const float*)d_in[27];
  const float* bt1 = (const float*)d_in[28];
  const float* g2  = (const float*)d_in[29];
  const float* bt2 = (const float*)d_in[30];
  const float* g3  = (const float*)d_in[31];
  const float* bt3 = (const float*)d_in[32];
  const float* g4  = (const float*)d_in[33];
  const float* bt4 = (const float*)d_in[34];

  float* ws   = (float*)d_ws;
  float* h1   = ws + OFF_H1;
  float* cout = ws + OFF_OUT;
  float* ta   = ws + OFF_TA;
  float* tb   = ws + OFF_TB;
  float* asrc = ws + OFF_ASRC;
  float* adst = ws + OFF_ADST;
  unsigned* mx = (unsigned*)(ws + OFF_MX);
  float* den  = ws + OFF_DEN;
  float* elog = ws + OFF_ELOG;
  float* wea1 = ws + OFF_WEA1;
  float* wea2 = ws + OFF_WEA2;
  float* bea1 = ws + OFF_BEA1;
  float* bea2 = ws + OFF_BEA2;
  float* bnm  = ws + OFF_BNM;
  float* bnv  = ws + OFF_BNV;
  float* outp = (float*)d_out;

  const dim3 blk128(128), blk256(256);

  // fold edge weights through attention vectors (once per launch)
  gat_wea_precompute<<<1, 256, 0, stream>>>(We1, ae1, be1, We2, ae2, be2,
                                            wea1, bea1, wea2, bea2);

  // ================= SparseConv1 =================
  {
    dim3 grid(cdivi(HC1, TBN), cdivi(N_NODES, TBM));
    gat_gemm_wmma_f32<<<grid, blk128, 0, stream>>>(x, W1, nullptr, nullptr, h1,
                                                   N_NODES, IN_F, HC1, 0);
  }
  gat_node_att<<<cdivi((long)N_NODES * 8, 256), blk256, 0, stream>>>(h1, as1, ad1, asrc, adst,
                                                                     N_NODES, C1_, HC1);
  gat_fill_f32<<<cdivi((long)N_NODES * 8, 256), blk256, 0, stream>>>((float*)mx, 0.f, (long)N_NODES * 8);
  gat_fill_f32<<<cdivi((long)N_NODES * 8, 256), blk256, 0, stream>>>(den, 0.f, (long)N_NODES * 8);
  gat_edge_logit<<<cdivi(N_EDGES, 256), blk256, 0, stream>>>(ea, src, dst, asrc, adst,
                                                             wea1, bea1, elog, mx, N_EDGES);
  gat_edge_ex<<<cdivi((long)N_EDGES * 8, 256), blk256, 0, stream>>>(elog, dst, mx, den, N_EDGES);
  gat_edge_alpha<<<cdivi((long)N_EDGES * 8, 256), blk256, 0, stream>>>(elog, dst, den, N_EDGES);
  gat_fill_f32<<<cdivi((long)N_NODES * HC1, 256), blk256, 0, stream>>>(cout, 0.f, (long)N_NODES * HC1);
  gat_scatter<<<8192, blk256, 0, stream>>>(h1, elog, src, dst, cout, N_EDGES, HC1, C1_);
  gat_bias_add<<<cdivi((long)N_NODES * HC1, 256), blk256, 0, stream>>>(cout, b1, (long)N_NODES * HC1, HC1);

  // fc1 + lrelu + residual(x), then BN1
  {
    dim3 grid(cdivi(IN_F, TBN), cdivi(N_NODES, TBM));
    gat_gemm_wmma_f32<<<grid, blk128, 0, stream>>>(cout, fc1_w, fc1_b, x, ta,
                                                   N_NODES, HC1, IN_F, 3);
  }
  gat_bn_stats<<<IN_F, blk256, 0, stream>>>(ta, bnm, bnv, N_NODES, IN_F);
  gat_bn_apply<<<cdivi((long)N_NODES * IN_F, 256), blk256, 0, stream>>>(ta, bnm, bnv, g1, bt1, N_NODES, IN_F);

  // fc2 + lrelu + residual(x), then BN2
  {
    dim3 grid(cdivi(IN_F, TBN), cdivi(N_NODES, TBM));
    gat_gemm_wmma_f32<<<grid, blk128, 0, stream>>>(ta, fc2_w, fc2_b, x, tb,
                                                   N_NODES, IN_F, IN_F, 3);
  }
  gat_bn_stats<<<IN_F, blk256, 0, stream>>>(tb, bnm, bnv, N_NODES, IN_F);
  gat_bn_apply<<<cdivi((long)N_NODES * IN_F, 256), blk256, 0, stream>>>(tb, bnm, bnv, g2, bt2, N_NODES, IN_F);

  // ================= SparseConv2 (input = tb) =================
  {
    dim3 grid(cdivi(HC2, TBN), cdivi(N_NODES, TBM));
    gat_gemm_wmma_f32<<<grid, blk128, 0, stream>>>(tb, W2, nullptr, nullptr, h1,
                                                   N_NODES, IN_F, HC2, 0);
  }
  gat_node_att<<<cdivi((long)N_NODES * 8, 256), blk256, 0, stream>>>(h1, as2, ad2, asrc, adst,
                                                                     N_NODES, C2_, HC2);
  gat_fill_f32<<<cdivi((long)N_NODES * 8, 256), blk256, 0, stream>>>((float*)mx, 0.f, (long)N_NODES * 8);
  gat_fill_f32<<<cdivi((long)N_NODES * 8, 256), blk256, 0, stream>>>(den, 0.f, (long)N_NODES * 8);
  gat_edge_logit<<<cdivi(N_EDGES, 256), blk256, 0, stream>>>(ea, src, dst, asrc, adst,
                                                             wea2, bea2, elog, mx, N_EDGES);
  gat_edge_ex<<<cdivi((long)N_EDGES * 8, 256), blk256, 0, stream>>>(elog, dst, mx, den, N_EDGES);
  gat_edge_alpha<<<cdivi((long)N_EDGES * 8, 256), blk256, 0, stream>>>(elog, dst, den, N_EDGES);
  gat_fill_f32<<<cdivi((long)N_NODES * HC2, 256), blk256, 0, stream>>>(cout, 0.f, (long)N_NODES * HC2);
  gat_scatter<<<8192, blk256, 0, stream>>>(h1, elog, src, dst, cout, N_EDGES, HC2, C2_);
  gat_bias_add<<<cdivi((long)N_NODES * HC2, 256), blk256, 0, stream>>>(cout, b2, (long)N_NODES * HC2, HC2);

  // fc3 + lrelu + residual(x), then BN3
  {
    dim3 grid(cdivi(IN_F, TBN), cdivi(N_NODES, TBM));
    gat_gemm_wmma_f32<<<grid, blk128, 0, stream>>>(cout, fc3_w, fc3_b, x, ta,
                                                   N_NODES, HC2, IN_F, 3);
  }
  gat_bn_stats<<<IN_F, blk256, 0, stream>>>(ta, bnm, bnv, N_NODES, IN_F);
  gat_bn_apply<<<cdivi((long)N_NODES * IN_F, 256), blk256, 0, stream>>>(ta, bnm, bnv, g3, bt3, N_NODES, IN_F);

  // fc4 + lrelu + residual(x), then BN4
  {
    dim3 grid(cdivi(IN_F, TBN), cdivi(N_NODES, TBM));
    gat_gemm_wmma_f32<<<grid, blk128, 0, stream>>>(ta, fc4_w, fc4_b, x, tb,
                                                   N_NODES, IN_F, IN_F, 3);
  }
  gat_bn_stats<<<IN_F, blk256, 0, stream>>>(tb, bnm, bnv, N_NODES, IN_F);
  gat_bn_apply<<<cdivi((long)N_NODES * IN_F, 256), blk256, 0, stream>>>(tb, bnm, bnv, g4, bt4, N_NODES, IN_F);

  // final projection
  gat_fc5<<<cdivi(N_NODES, 8), blk256, 0, stream>>>(tb, fc5_w, fc5_b, outp, N_NODES, IN_F);
}